// PtModule_76166950027787
// MI455X (gfx1250) — compile-verified
//
#include <hip/hip_runtime.h>

// y = x*2 + 2 over 2^27 f32 (1 GiB traffic). 100% HBM-bound: floor ~46us at
// 23.3 TB/s. Optimal gfx1250 path = b128 global load/store with TH=NT (stream
// is 5x the 192MB L2), wave32 blocks, max MLP. WMMA intentionally unused.

typedef float v4f __attribute__((ext_vector_type(4)));

#define BLOCK 256u   // 8 wave32s per block
#define VPT   4u     // float4 chunks per thread -> 4 b128 NT loads in flight

__global__ __launch_bounds__(BLOCK) void scale_add_kernel(
    const float* __restrict__ x, float* __restrict__ y,
    unsigned n4, unsigned rem) {
  const unsigned tid        = threadIdx.x;
  const unsigned block_base = blockIdx.x * (BLOCK * VPT);  // in float4 units

  const v4f* __restrict__ xi = (const v4f*)x;
  v4f* __restrict__ yo = (v4f*)y;

  if (block_base + BLOCK * VPT <= n4) {
    // ---- Fast path (all blocks for this shape): no predication at all ----
    v4f r[VPT];
#pragma unroll
    for (unsigned k = 0; k < VPT; ++k)
      r[k] = __builtin_nontemporal_load(&xi[block_base + k * BLOCK + tid]);
#pragma unroll
    for (unsigned k = 0; k < VPT; ++k) {
      v4f v = r[k] * 2.0f + 2.0f;
      __builtin_nontemporal_store(v, &yo[block_base + k * BLOCK + tid]);
    }
  } else {
    // ---- Guarded edge path (dead for this shape, kept shape-generic) ----
#pragma unroll
    for (unsigned k = 0; k < VPT; ++k) {
      const unsigned idx = block_base + k * BLOCK + tid;
      if (idx < n4) {
        v4f v = __builtin_nontemporal_load(&xi[idx]) * 2.0f + 2.0f;
        __builtin_nontemporal_store(v, &yo[idx]);
      }
    }
    // Scalar tail (n % 4 != 0) handled by the last block's first threads.
    if (rem && blockIdx.x == gridDim.x - 1 && tid < rem) {
      const unsigned i = n4 * 4u + tid;
      __builtin_nontemporal_store(
          __builtin_nontemporal_load(&x[i]) * 2.0f + 2.0f, &y[i]);
    }
  }
}

extern "C" void kernel_launch(void* const* d_in, const int* in_sizes, int n_in,
                              void* d_out, int out_size, void* d_ws, size_t ws_size,
                              hipStream_t stream) {
  const float* x = (const float*)d_in[0];
  float* y = (float*)d_out;

  const unsigned n   = (unsigned)in_sizes[0];  // 2^27 for this shape
  const unsigned n4  = n >> 2;
  const unsigned rem = n & 3u;

  const unsigned per_block = BLOCK * VPT;                   // float4s per block
  unsigned grid = (n4 + per_block - 1) / per_block;         // 32768 here
  if (grid == 0) grid = 1;

  scale_add_kernel<<<grid, BLOCK, 0, stream>>>(x, y, n4, rem);
}